// HamiltonAdam_24446953849133
// MI455X (gfx1250) — compile-verified
//
#include <hip/hip_runtime.h>
#include <hip/hip_bf16.h>

// Hamilton–Adams RGGB demosaic, fused single pass.
// MI455X reasoning: memory/VALU-bound sparse stencil (~302 MB/call -> ~13us HBM
// floor @23.3TB/s, ~1 GFLOP fp32). WMMA f32 is 16x16x4 dense only -> useless for
// 3-tap sparse stencils. We instead use the CDNA5 async global->LDS path
// (global_load_async_to_lds_b32 + s_wait_asynccnt), big LDS tiles (320KB/WGP),
// wave32-uniform parity branching, and nontemporal float2 stores.

#define TX 64
#define TY 32
#define RAW_W (TX + 6)   // 70
#define RAW_H (TY + 6)   // 38
#define GRN_W (TX + 2)   // 66
#define GRN_H (TY + 2)   // 34
#define NTHR 256

typedef float __attribute__((ext_vector_type(2))) f2;

// ---- CDNA5 async global->LDS (ASYNCcnt) ------------------------------------
__device__ __forceinline__ void async_g2l_b32(const float* gp, float* lp) {
#if __has_builtin(__builtin_amdgcn_global_load_async_to_lds_b32)
  typedef __attribute__((address_space(1))) int GI;  // global int*
  typedef __attribute__((address_space(3))) int LI;  // LDS int*
  __builtin_amdgcn_global_load_async_to_lds_b32((GI*)gp, (LI*)lp, 0, 0);
#else
  unsigned loff = (unsigned)(unsigned long long)lp;  // low 32b of flat LDS addr = LDS offset
  asm volatile("global_load_async_to_lds_b32 %0, %1, off" :: "v"(loff), "v"(gp) : "memory");
#endif
}

__device__ __forceinline__ void wait_async0() {
#if __has_builtin(__builtin_amdgcn_s_wait_asynccnt)
  __builtin_amdgcn_s_wait_asynccnt(0);
#else
  asm volatile("s_wait_asynccnt 0x0" ::: "memory");
#endif
}

__device__ __forceinline__ float sgn_blend(float d, float pos, float neg) {
  // ((1+sign(d))*pos + (1-sign(d))*neg) * 0.5
  float loc = (d > 0.f) ? 1.f : ((d < 0.f) ? -1.f : 0.f);
  return 0.5f * ((1.f + loc) * pos + (1.f - loc) * neg);
}

template<bool EDGE>
__device__ __forceinline__ void demosaic_block(
    const float* __restrict__ xin, float* __restrict__ out,
    float* sraw, float* sgrn, int Hh, int Wh, int b, int by0, int bx0) {
  const int H = Hh * 2, W = Wh * 2;
  const int oy = by0 - 3, ox = bx0 - 3;
  const int tid = threadIdx.x;
  const size_t inBase = (size_t)b * 4 * Hh * Wh;

  // ---- Stage 1: async-gather full-res Bayer tile (replicate-clamped) to LDS
  for (int i = tid; i < RAW_H * RAW_W; i += NTHR) {
    int ly = i / RAW_W;
    int lx = i - ly * RAW_W;
    int gy = oy + ly, gx = ox + lx;
    if (EDGE) {
      gy = gy < 0 ? 0 : (gy >= H ? H - 1 : gy);
      gx = gx < 0 ? 0 : (gx >= W ? W - 1 : gx);
    }
    int c = ((gy & 1) << 1) | (gx & 1);  // rggb sub-plane
    const float* gp = xin + inBase + ((size_t)c * Hh + (gy >> 1)) * Wh + (gx >> 1);
    async_g2l_b32(gp, &sraw[i]);
  }
  wait_async0();
  __syncthreads();

  // ---- Stage 2: green plane for tile + 1 halo (algo1) ----
  for (int i = tid; i < GRN_H * GRN_W; i += NTHR) {
    int ly = i / GRN_W;
    int lx = i - ly * GRN_W;
    int gy = by0 - 1 + ly, gx = bx0 - 1 + lx;
    if (EDGE) {  // halo green must be green(clamped coord), not green at -1
      gy = gy < 0 ? 0 : (gy >= H ? H - 1 : gy);
      gx = gx < 0 ? 0 : (gx >= W ? W - 1 : gx);
    }
    const float* r = &sraw[(gy - oy) * RAW_W + (gx - ox)];
    float c00 = r[0];
    float g;
    if (((gy ^ gx) & 1) != 0) {
      g = c00;  // green site: pass-through
    } else {
      float xm1 = r[-1], xp1 = r[1];
      float ym1 = r[-RAW_W], yp1 = r[RAW_W];
      float c2 = r[-2] + r[2] - 2.f * c00;
      float c3 = r[-2 * RAW_W] + r[2 * RAW_W] - 2.f * c00;
      float rawh = 0.5f * (xm1 + xp1) - 0.25f * c2;
      float rawv = 0.5f * (ym1 + yp1) - 0.25f * c3;
      float CLh = fabsf(xm1 - xp1) + fabsf(c2);
      float CLv = fabsf(ym1 - yp1) + fabsf(c3);
      g = sgn_blend(CLh - CLv, rawv, rawh);
    }
    sgrn[i] = g;
  }
  __syncthreads();

  // ---- Stage 3: red/blue (algo2) + stores. One wave per 64px row. ----
  float* outR = out + ((size_t)b * 3 + 0) * (size_t)H * W;
  float* outG = out + ((size_t)b * 3 + 1) * (size_t)H * W;
  float* outB = out + ((size_t)b * 3 + 2) * (size_t)H * W;

  const int pairX = tid & 31;
  const int rowGrp = tid >> 5;  // wave id -> row within group of 8

  // masked-plane read: clamp + Bayer-parity mask only needed on border blocks
  auto MR = [&](int y, int x, int py, int px) -> float {
    int cy = y, cx = x;
    if (EDGE) {
      cy = cy < 0 ? 0 : (cy >= H ? H - 1 : cy);
      cx = cx < 0 ? 0 : (cx >= W ? W - 1 : cx);
    }
    float v = sraw[(cy - oy) * RAW_W + (cx - ox)];
    if (EDGE) {
      if ((((cy ^ py) | (cx ^ px)) & 1) != 0) v = 0.0f;
    }
    return v;
  };

  for (int p = 0; p < TY / (NTHR / 32); ++p) {
    const int row = p * (NTHR / 32) + rowGrp;
    const int gy = by0 + row;
    const int gxA = bx0 + 2 * pairX;   // even column pixel
    const int gxB = gxA + 1;           // odd column pixel

    // green 3x4 neighborhood covering the pixel pair
    const float* gc = &sgrn[(row + 1) * GRN_W + (2 * pairX + 1)];
    float gA = gc[0], gB = gc[1];
    float gAxm = gc[-1], gBxp = gc[2];
    float gAym = gc[-GRN_W], gBym = gc[-GRN_W + 1];
    float gAyp = gc[GRN_W],  gByp = gc[GRN_W + 1];
    float gAmm = gc[-GRN_W - 1], gBmp = gc[-GRN_W + 2];
    float gApm = gc[GRN_W - 1],  gBpp = gc[GRN_W + 2];
    float gAmp = gBym, gApp = gByp, gBmm = gAym, gBpm = gAyp;
    float gAxp = gB, gBxm = gA;

    float cg0A = 0.25f * (gAxm + gAxp) - 0.5f * gA;
    float cg1A = 0.25f * (gAym + gAyp) - 0.5f * gA;
    float cg2A = gAmm + gApp - 2.f * gA;
    float cg3A = gAmp + gApm - 2.f * gA;
    float cg0B = 0.25f * (gBxm + gBxp) - 0.5f * gB;
    float cg1B = 0.25f * (gBym + gByp) - 0.5f * gB;
    float cg2B = gBmm + gBpp - 2.f * gB;
    float cg3B = gBmp + gBpm - 2.f * gB;

    const int rrA = (row + 3) * RAW_W + (2 * pairX + 3);
    float rawA = sraw[rrA];
    float rawB = sraw[rrA + 1];

    float redA, bluA, redB, bluB;
    if ((gy & 1) == 0) {
      // A = R site, B = Gr site
      redA = rawA;
      { // blue at R: diagonal B-plane chan
        float mm = MR(gy - 1, gxA - 1, 1, 1), pp = MR(gy + 1, gxA + 1, 1, 1);
        float mp = MR(gy - 1, gxA + 1, 1, 1), pm = MR(gy + 1, gxA - 1, 1, 1);
        float Cp = 0.5f * (mm + pp) - 0.25f * cg2A;
        float Cn = 0.5f * (mp + pm) - 0.25f * cg3A;
        float CLp = fabsf(pp - mm) + fabsf(cg2A);
        float CLn = fabsf(pm - mp) + fabsf(cg3A);
        bluA = sgn_blend(CLp - CLn, Cn, Cp);
      }
      redB = 0.5f * (rawA + MR(gy, gxB + 1, 0, 0)) - cg0B;                 // Ch (R plane)
      bluB = 0.5f * (MR(gy - 1, gxB, 1, 1) + MR(gy + 1, gxB, 1, 1)) - cg1B; // Cv (B plane)
    } else {
      // A = Gb site, B = B site
      redA = 0.5f * (MR(gy - 1, gxA, 0, 0) + MR(gy + 1, gxA, 0, 0)) - cg1A; // Cv (R plane)
      bluA = 0.5f * (MR(gy, gxA - 1, 1, 1) + rawB) - cg0A;                   // Ch (B plane)
      { // red at B: diagonal R-plane chan
        float mm = MR(gy - 1, gxB - 1, 0, 0), pp = MR(gy + 1, gxB + 1, 0, 0);
        float mp = MR(gy - 1, gxB + 1, 0, 0), pm = MR(gy + 1, gxB - 1, 0, 0);
        float Cp = 0.5f * (mm + pp) - 0.25f * cg2B;
        float Cn = 0.5f * (mp + pm) - 0.25f * cg3B;
        float CLp = fabsf(pp - mm) + fabsf(cg2B);
        float CLn = fabsf(pm - mp) + fabsf(cg3B);
        redB = sgn_blend(CLp - CLn, Cn, Cp);
      }
      bluB = rawB;
    }

    const size_t off = (size_t)gy * W + gxA;
    f2 vr = {redA, redB};
    f2 vg = {gA, gB};
    f2 vb = {bluA, bluB};
    __builtin_nontemporal_store(vr, (f2*)(outR + off));
    __builtin_nontemporal_store(vg, (f2*)(outG + off));
    __builtin_nontemporal_store(vb, (f2*)(outB + off));
  }
}

__global__ __launch_bounds__(NTHR) void demosaic_kernel(
    const float* __restrict__ xin, float* __restrict__ out, int Hh, int Wh) {
  __shared__ float sraw[RAW_H * RAW_W];   // 38*70*4  = 10640 B
  __shared__ float sgrn[GRN_H * GRN_W];   // 34*66*4  =  8976 B
  const int b = blockIdx.z;
  const int by0 = blockIdx.y * TY;
  const int bx0 = blockIdx.x * TX;
  const int H = Hh * 2, W = Wh * 2;
  const bool edge = (by0 == 0) || (bx0 == 0) || (by0 + TY >= H) || (bx0 + TX >= W);
  if (edge)
    demosaic_block<true>(xin, out, sraw, sgrn, Hh, Wh, b, by0, bx0);
  else
    demosaic_block<false>(xin, out, sraw, sgrn, Hh, Wh, b, by0, bx0);
}

extern "C" void kernel_launch(void* const* d_in, const int* in_sizes, int n_in,
                              void* d_out, int out_size, void* d_ws, size_t ws_size,
                              hipStream_t stream) {
  (void)n_in; (void)out_size; (void)d_ws; (void)ws_size;
  const float* x = (const float*)d_in[0];
  float* out = (float*)d_out;
  const int Hh = 768, Wh = 768;                 // reference: x [B,4,768,768]
  const int B = in_sizes[0] / (4 * Hh * Wh);
  dim3 grid((2 * Wh) / TX, (2 * Hh) / TY, B);   // 24 x 48 x 8
  demosaic_kernel<<<grid, NTHR, 0, stream>>>(x, out, Hh, Wh);
}